// FPGA_GNN_18511309046119
// MI455X (gfx1250) — compile-verified
//
#include <hip/hip_runtime.h>
#include <hip/hip_bf16.h>
#include <stdint.h>

#define N_NODES  100000
#define N_EDGES  800000
#define N_GRAPHS 512
#define IN_DIM   32
#define HID      128
#define MLP_H    64
#define N_TGT    6

typedef __attribute__((ext_vector_type(2))) float v2f;
typedef __attribute__((ext_vector_type(8))) float v8f;

// ---------------------------------------------------------------------------
// Edge scatter-add: agg[dst[e], :] += feat[src[e], :]
// One thread per (edge, 4 dims): float4 (global_load_b128) feature reads,
// 4x global_atomic_add_f32. Dim-major so loads/atomics coalesce per wave.
// ---------------------------------------------------------------------------
__global__ void __launch_bounds__(256)
scatter_add_kernel(const float* __restrict__ feat,
                   const int*   __restrict__ src,
                   const int*   __restrict__ dst,
                   float* __restrict__ agg,
                   int n_edges, int log2dim) {
  const int dim   = 1 << log2dim;
  const int log2q = log2dim - 2;              // quads per row = dim/4
  const long long total  = (long long)n_edges << log2q;
  const long long stride = (long long)gridDim.x * blockDim.x;
  for (long long i = (long long)blockIdx.x * blockDim.x + threadIdx.x;
       i < total; i += stride) {
    const int e = (int)(i >> log2q);
    const int j = ((int)i & ((1 << log2q) - 1)) << 2;  // dim offset
    const int s = src[e];
    const int t = dst[e];
    const float4 v = *(const float4*)(feat + (size_t)s * dim + j);
    float* ap = agg + (size_t)t * dim + j;
    atomicAdd(ap + 0, v.x);
    atomicAdd(ap + 1, v.y);
    atomicAdd(ap + 2, v.z);
    atomicAdd(ap + 3, v.w);
  }
}

// ---------------------------------------------------------------------------
// Fused SAGE linear:  out = relu(Aagg @ Wl^T + bias + Aself @ Wr^T)
//
// Per block: stage Wl,Wr into LDS once (layer1: 128 KB of the 320 KB WGP LDS).
// Per wave32: a 32-row x 128-col output tile = 2 M-tiles x 8 N-tiles of
// V_WMMA_F32_16X16X4_F32; each B fragment (ds_load_b64 from LDS) feeds two
// WMMAs, each A fragment (global v2f) feeds eight.
//
// Fragment layouts per ISA 7.12.2 (32-bit A 16x4): lanes 0-15 carry {K,K+1}
// of row M=lane, lanes 16-31 carry {K+2,K+3} of row M=lane-16. B mirrors it
// from row-major W[HID][K] (B[k][n] = W[n][k]). D: VGPR v <-> M = v +
// 8*(lane>=16), N = lane&15; bias+ReLU fused on store.
// M % 32 == 0 and the tail guard is wave-uniform -> EXEC all-ones at WMMA.
// ---------------------------------------------------------------------------
template <int K>
__global__ void __launch_bounds__(256)
sage_linear_relu_wmma(const float* __restrict__ Aagg,   // [M,K] aggregated msgs
                      const float* __restrict__ Wl,     // [HID,K]
                      const float* __restrict__ Aself,  // [M,K] self features
                      const float* __restrict__ Wr,     // [HID,K]
                      const float* __restrict__ bias,   // [HID]
                      float* __restrict__ out,          // [M,HID]
                      int M) {
  __shared__ float sW[2][HID * K];

  // Cooperative weight staging (float4 copies).
  {
    const float4* gl = (const float4*)Wl;
    const float4* gr = (const float4*)Wr;
    float4* sl = (float4*)&sW[0][0];
    float4* sr = (float4*)&sW[1][0];
    const int n4 = HID * K / 4;
    for (int i = threadIdx.x; i < n4; i += 256) {
      sl[i] = gl[i];
      sr[i] = gr[i];
    }
  }
  __syncthreads();

  const int lane = threadIdx.x & 31;
  const int wave = (int)((blockIdx.x * blockDim.x + threadIdx.x) >> 5);
  const int row0 = wave * 32;
  if (row0 >= M) return;  // wave-uniform; no barriers after this point

  const int mrow0 = row0 + (lane & 15);   // A rows this lane feeds
  const int mrow1 = mrow0 + 16;
  const int khalf = (lane >> 4) * 2;      // lanes 16-31 carry K+2,K+3
  const int ncol  = lane & 15;            // column within 16-wide N tile

  v8f acc[2][8];
#pragma unroll
  for (int mt = 0; mt < 2; ++mt)
#pragma unroll
    for (int nt = 0; nt < 8; ++nt)
      acc[mt][nt] = (v8f){0.f, 0.f, 0.f, 0.f, 0.f, 0.f, 0.f, 0.f};

#pragma unroll
  for (int pass = 0; pass < 2; ++pass) {
    const float* __restrict__ A = pass ? Aself : Aagg;
    const float* arow0 = A + (size_t)mrow0 * K + khalf;
    const float* arow1 = A + (size_t)mrow1 * K + khalf;
#pragma unroll 4
    for (int k = 0; k < K; k += 4) {
      const v2f a0 = *(const v2f*)(arow0 + k);
      const v2f a1 = *(const v2f*)(arow1 + k);
#pragma unroll
      for (int nt = 0; nt < 8; ++nt) {
        const v2f b = *(const v2f*)&sW[pass][(nt * 16 + ncol) * K + k + khalf];
        acc[0][nt] = __builtin_amdgcn_wmma_f32_16x16x4_f32(
            false, a0, false, b, (short)0, acc[0][nt], false, false);
        acc[1][nt] = __builtin_amdgcn_wmma_f32_16x16x4_f32(
            false, a1, false, b, (short)0, acc[1][nt], false, false);
      }
    }
  }

  // Fused bias + ReLU + store per ISA D-matrix layout.
  const int mbase = row0 + ((lane >> 4) << 3);
#pragma unroll
  for (int mt = 0; mt < 2; ++mt) {
    const int mb = mbase + mt * 16;
#pragma unroll
    for (int nt = 0; nt < 8; ++nt) {
      const float bv = bias[nt * 16 + ncol];
#pragma unroll
      for (int v = 0; v < 8; ++v) {
        float val = acc[mt][nt][v] + bv;
        val = fmaxf(val, 0.0f);
        out[(size_t)(mb + v) * HID + nt * 16 + ncol] = val;
      }
    }
  }
}

// ---------------------------------------------------------------------------
// Global add pool: pooled[batch[n], d] += h[n, d]
// ---------------------------------------------------------------------------
__global__ void __launch_bounds__(256)
pool_kernel(const float* __restrict__ h,
            const int*   __restrict__ batch,
            float* __restrict__ pooled) {
  const long long total  = (long long)N_NODES * HID;
  const long long stride = (long long)gridDim.x * blockDim.x;
  for (long long i = (long long)blockIdx.x * blockDim.x + threadIdx.x;
       i < total; i += stride) {
    const int n = (int)(i >> 7);
    const int d = (int)i & (HID - 1);
    atomicAdd(&pooled[(size_t)batch[n] * HID + d], h[i]);
  }
}

// ---------------------------------------------------------------------------
// MLP head: [HID+1] -> 64 -> 32 -> 6 per graph, one 64-thread block per graph.
// Tiny (512 x ~21 KFLOP): plain VALU with LDS staging of activations.
// ---------------------------------------------------------------------------
__global__ void __launch_bounds__(64)
mlp_head_kernel(const float* __restrict__ pooled,    // [G, HID]
                const float* __restrict__ strategy,  // [G, 1]
                const float* __restrict__ Wm1, const float* __restrict__ bm1,
                const float* __restrict__ Wm2, const float* __restrict__ bm2,
                const float* __restrict__ Wm3, const float* __restrict__ bm3,
                float* __restrict__ out) {           // [G, N_TGT]
  const int g = blockIdx.x;
  const int t = threadIdx.x;
  __shared__ float zin[HID + 1];
  __shared__ float z1[MLP_H];
  __shared__ float z2[MLP_H / 2];

  for (int d = t; d < HID; d += 64) zin[d] = pooled[(size_t)g * HID + d];
  if (t == 0) zin[HID] = strategy[g];
  __syncthreads();

  {
    float s = bm1[t];
    const float* w = Wm1 + (size_t)t * (HID + 1);
    for (int k = 0; k < HID + 1; ++k) s += w[k] * zin[k];
    z1[t] = fmaxf(s, 0.0f);
  }
  __syncthreads();

  if (t < MLP_H / 2) {
    float s = bm2[t];
    const float* w = Wm2 + (size_t)t * MLP_H;
    for (int k = 0; k < MLP_H; ++k) s += w[k] * z1[k];
    z2[t] = fmaxf(s, 0.0f);
  }
  __syncthreads();

  if (t < N_TGT) {
    float s = bm3[t];
    const float* w = Wm3 + (size_t)t * (MLP_H / 2);
    for (int k = 0; k < MLP_H / 2; ++k) s += w[k] * z2[k];
    out[(size_t)g * N_TGT + t] = s;
  }
}

// ---------------------------------------------------------------------------
extern "C" void kernel_launch(void* const* d_in, const int* in_sizes, int n_in,
                              void* d_out, int out_size, void* d_ws, size_t ws_size,
                              hipStream_t stream) {
  (void)in_sizes; (void)n_in; (void)out_size; (void)ws_size;

  const float* x        = (const float*)d_in[0];
  const int*   edge     = (const int*)  d_in[1];   // [2, N_EDGES] flat
  const int*   batch    = (const int*)  d_in[2];
  const float* strategy = (const float*)d_in[3];
  const float* Wl0 = (const float*)d_in[4];
  const float* bl0 = (const float*)d_in[5];
  const float* Wr0 = (const float*)d_in[6];
  const float* Wl1 = (const float*)d_in[7];
  const float* bl1 = (const float*)d_in[8];
  const float* Wr1 = (const float*)d_in[9];
  const float* Wm1 = (const float*)d_in[10];
  const float* bm1 = (const float*)d_in[11];
  const float* Wm2 = (const float*)d_in[12];
  const float* bm2 = (const float*)d_in[13];
  const float* Wm3 = (const float*)d_in[14];
  const float* bm3 = (const float*)d_in[15];

  const int* src = edge;
  const int* dst = edge + N_EDGES;

  // Workspace layout (bytes): h0 | agg (shared by layer0/1) | h1 | pooled
  const size_t nodeMat = (size_t)N_NODES * HID * sizeof(float);  // 51.2 MB
  char*  ws     = (char*)d_ws;
  float* h0     = (float*)(ws);
  float* agg    = (float*)(ws + nodeMat);
  float* h1     = (float*)(ws + 2 * nodeMat);
  float* pooled = (float*)(ws + 3 * nodeMat);

  // 8 waves/block, 32 rows/wave -> 256 rows/block; N_NODES = 3125 tiles of 32.
  const int sageBlocks = ((N_NODES + 31) / 32 + 7) / 8;

  // -------- Layer 0 --------
  hipMemsetAsync(agg, 0, (size_t)N_NODES * IN_DIM * sizeof(float), stream);
  scatter_add_kernel<<<4096, 256, 0, stream>>>(x, src, dst, agg, N_EDGES, 5);
  sage_linear_relu_wmma<IN_DIM><<<sageBlocks, 256, 0, stream>>>(
      agg, Wl0, x, Wr0, bl0, h0, N_NODES);

  // -------- Layer 1 --------
  hipMemsetAsync(agg, 0, nodeMat, stream);
  scatter_add_kernel<<<8192, 256, 0, stream>>>(h0, src, dst, agg, N_EDGES, 7);
  sage_linear_relu_wmma<HID><<<sageBlocks, 256, 0, stream>>>(
      agg, Wl1, h0, Wr1, bl1, h1, N_NODES);

  // -------- Pool + head --------
  hipMemsetAsync(pooled, 0, (size_t)N_GRAPHS * HID * sizeof(float), stream);
  pool_kernel<<<4096, 256, 0, stream>>>(h1, batch, pooled);
  mlp_head_kernel<<<N_GRAPHS, 64, 0, stream>>>(
      pooled, strategy, Wm1, bm1, Wm2, bm2, Wm3, bm3, (float*)d_out);
}